// Squre02_05_20684562497583
// MI455X (gfx1250) — compile-verified
//
#include <hip/hip_runtime.h>
#include <stdint.h>

// Elementwise 16-step spiking-neuron scan, fwd only.
// Memory-bound: 537 MB traffic -> ~23us floor @ 23.3 TB/s.
// Division removed: sign((v-T)/(|v|+1)) == sign(v-T) since denom > 0.
// Data path: async global->LDS B128 double-buffer (ASYNCcnt) + ds_load_b128
// + non-temporal B128 stores. Coefficients land in SGPRs (uniform loads).

typedef __attribute__((ext_vector_type(4))) float v4f;

#define TPB 256

__device__ __forceinline__ float neuron16(float xin,
                                          const float* __restrict__ h,
                                          const float* __restrict__ d,
                                          const float* __restrict__ T)
{
    float v = __builtin_fabsf(xin);
    float z = 0.0f;
    float o = 0.0f;
#pragma unroll
    for (int t = 0; t < 16; ++t) {
        v = __builtin_fmaf(z, -h[t], v);   // v -= z*h[t]  (neg folds into FMA)
        bool fire = v > T[t];              // == ((v-T)/(|v|+1) > 0)
        z = fire ? 1.0f : 0.0f;
        o = __builtin_fmaf(z, d[t], o);    // out += z*d[t]
    }
    float s = (xin > 0.0f) ? 1.0f : ((xin < 0.0f) ? -1.0f : 0.0f);
    return o * s;                          // sign(x), with sign(0)=0
}

__global__ __launch_bounds__(TPB) void snn_main(const float* __restrict__ x,
                                                const float* __restrict__ hp,
                                                const float* __restrict__ dp,
                                                const float* __restrict__ Tp,
                                                float* __restrict__ out,
                                                int ntiles)
{
    __shared__ v4f smem[2][TPB];
    const int tid = threadIdx.x;

    // Uniform coefficient loads -> SGPRs after full unroll.
    float h[16], d[16], T[16];
#pragma unroll
    for (int t = 0; t < 16; ++t) { h[t] = hp[t]; d[t] = dp[t]; T[t] = Tp[t]; }

    int tile = blockIdx.x;
    const int stride = gridDim.x;
    if (tile >= ntiles) return;

    const unsigned lds0 = (unsigned)(uintptr_t)&smem[0][tid];
    const unsigned lds1 = (unsigned)(uintptr_t)&smem[1][tid];

    // Prologue: async-prefetch first tile into buffer 0.
    {
        unsigned off = ((unsigned)(tile * TPB + tid)) * 16u;
        asm volatile("global_load_async_to_lds_b128 %0, %1, %2"
                     :: "v"(lds0), "v"(off), "s"(x) : "memory");
    }

    int buf = 0;
    for (; tile < ntiles; tile += stride, buf ^= 1) {
        const int next = tile + stride;
        if (next < ntiles) {
            // Prefetch next tile into the other buffer, then wait for current.
            unsigned off = ((unsigned)(next * TPB + tid)) * 16u;
            unsigned lds = buf ? lds0 : lds1;
            asm volatile("global_load_async_to_lds_b128 %0, %1, %2"
                         :: "v"(lds), "v"(off), "s"(x) : "memory");
            asm volatile("s_wait_asynccnt 0x1" ::: "memory");
        } else {
            asm volatile("s_wait_asynccnt 0x0" ::: "memory");
        }

        // Each lane reads back exactly the 16B it requested -> no barrier needed.
        v4f xv = smem[buf][tid];

        v4f r;
        r.x = neuron16(xv.x, h, d, T);
        r.y = neuron16(xv.y, h, d, T);
        r.z = neuron16(xv.z, h, d, T);
        r.w = neuron16(xv.w, h, d, T);

        v4f* o4 = (v4f*)out;
        __builtin_nontemporal_store(r, &o4[tile * TPB + tid]);
    }
}

// Scalar cleanup for elements not covered by full 256x-float4 tiles
// (empty for the 64x1024x1024 reference shape).
__global__ void snn_tail(const float* __restrict__ x,
                         const float* __restrict__ hp,
                         const float* __restrict__ dp,
                         const float* __restrict__ Tp,
                         float* __restrict__ out,
                         long start, long n)
{
    long i = start + (long)blockIdx.x * blockDim.x + threadIdx.x;
    if (i >= n) return;
    float h[16], d[16], T[16];
#pragma unroll
    for (int t = 0; t < 16; ++t) { h[t] = hp[t]; d[t] = dp[t]; T[t] = Tp[t]; }
    out[i] = neuron16(x[i], h, d, T);
}

extern "C" void kernel_launch(void* const* d_in, const int* in_sizes, int n_in,
                              void* d_out, int out_size, void* d_ws, size_t ws_size,
                              hipStream_t stream) {
    const float* x  = (const float*)d_in[0];
    const float* h  = (const float*)d_in[1];
    const float* dd = (const float*)d_in[2];
    const float* T  = (const float*)d_in[3];
    float* out = (float*)d_out;

    const long n     = (long)in_sizes[0];
    const long nvec  = n / 4;
    const int ntiles = (int)(nvec / TPB);

    if (ntiles > 0) {
        int blocks = ntiles < 8192 ? ntiles : 8192;   // ~8 tiles/block pipeline depth
        snn_main<<<blocks, TPB, 0, stream>>>(x, h, dd, T, out, ntiles);
    }
    const long done = (long)ntiles * TPB * 4;
    const long rem  = n - done;
    if (rem > 0) {
        int tb = (int)((rem + TPB - 1) / TPB);
        snn_tail<<<tb, TPB, 0, stream>>>(x, h, dd, T, out, done, n);
    }
}